// FingerNet_79293686219252
// MI455X (gfx1250) — compile-verified
//
#include <hip/hip_runtime.h>
#include <cstdint>

// ---------------- problem geometry ----------------
#define B_IMG   32
#define HF      512
#define WF      512
#define IMG     (HF*WF)            // 262144
#define HL      64
#define WL      64
#define MAPPX   (HL*WL)            // 4096
#define THRESH  0.45f

// ---------------- output layout (flat f32, return order) ----------------
#define OFF_ENH    0
#define OFF_SEGBIG (B_IMG*IMG)                  // 8388608
#define OFF_CO     (2*B_IMG*IMG)                // 16777216
#define OFF_CC     (OFF_CO + B_IMG*MAPPX)       // 16908288
#define OFF_MNT    (OFF_CC + B_IMG*MAPPX)       // 17039360
#define OFF_CONF   (OFF_MNT + B_IMG)            // 17039392

// ---------------- CDNA5 async global->LDS DMA helpers ----------------
// GVS mode: each lane copies 16B from global base saddr plus offset vaddr to the LDS address in vdst. Tracked by ASYNCcnt;
// wave waits with s_wait_asynccnt, then a workgroup barrier publishes the tile.
__device__ __forceinline__ void async_b128(unsigned lds_byte_off,
                                           const void* gbase,
                                           unsigned gbyte_off) {
  asm volatile("global_load_async_to_lds_b128 %0, %1, %2"
               :
               : "v"(lds_byte_off), "v"(gbyte_off), "s"(gbase)
               : "memory");
}

__device__ __forceinline__ void wait_async0() {
#if __has_builtin(__builtin_amdgcn_s_wait_asynccnt)
  __builtin_amdgcn_s_wait_asynccnt(0);
#else
  asm volatile("s_wait_asynccnt 0" ::: "memory");
#endif
}

// Low 32 bits of a generic pointer to LDS are the LDS byte offset
// (ISA 10.2, LDS aperture).
__device__ __forceinline__ unsigned lds_off_of(const void* p) {
  return (unsigned)(uintptr_t)p;
}

// ---------------- kernel 1: per-image min/max of enhanceImageP ----------------
// One block per image, 32 waves, b128 loads: 64 float4 per thread.
__global__ __launch_bounds__(1024) void kminmax(const float* __restrict__ enh,
                                                float* __restrict__ mm) {
  __shared__ float smin[1024];
  __shared__ float smax[1024];
  const int b = blockIdx.x, t = threadIdx.x;
  const float4* p = (const float4*)(enh + (size_t)b * IMG);   // 65536 float4
  float lo =  __builtin_inff();
  float hi = -__builtin_inff();
#pragma unroll 4
  for (int j = 0; j < 64; ++j) {
    int i = t + j * 1024;
    __builtin_prefetch(p + i + 4096, 0, 1);    // global_prefetch_b8
    float4 v = p[i];
    lo = fminf(lo, fminf(fminf(v.x, v.y), fminf(v.z, v.w)));
    hi = fmaxf(hi, fmaxf(fmaxf(v.x, v.y), fmaxf(v.z, v.w)));
  }
  smin[t] = lo; smax[t] = hi;
  __syncthreads();
  for (int s = 512; s > 0; s >>= 1) {
    if (t < s) {
      smin[t] = fminf(smin[t], smin[t + s]);
      smax[t] = fmaxf(smax[t], smax[t + s]);
    }
    __syncthreads();
  }
  if (t == 0) { mm[b] = smin[0]; mm[B_IMG + b] = smax[0]; }
}

// ---------------- kernel 2: enh_norm (vectorized) ----------------
__global__ __launch_bounds__(256) void knorm(const float* __restrict__ enh,
                                             const float* __restrict__ mm,
                                             float* __restrict__ out) {
  const int i4 = blockIdx.x * 256 + threadIdx.x;   // < 2097152 float4
  const int b  = i4 >> 16;                         // 65536 float4 per image
  const float mi = mm[b], mx = mm[B_IMG + b];
  const float den = mx - mi + 1e-6f;               // exact reference denom
  float4 v = ((const float4*)enh)[i4];
  float4 r = make_float4((v.x - mi) / den, (v.y - mi) / den,
                         (v.z - mi) / den, (v.w - mi) / den);
  ((float4*)out)[i4] = r;
}

// ---------------- kernel 3: fused 64x64 map pipeline ----------------
// dilate(round(seg),5) -> segment + x8 nearest upsample store
// cm = c*segment ; 15x15 NMS ; confO/confC ; per-image keep count
__global__ __launch_bounds__(256) void kmaps(const float* __restrict__ seg,
                                             const float* __restrict__ c,
                                             const float* __restrict__ o,
                                             float* __restrict__ out) {
  __shared__ float sA[MAPPX];   // raw seg -> rounded seg -> cm
  __shared__ float sH[MAPPX];   // separable-max scratch
  __shared__ float sC[MAPPX];   // c tile; reused for count reduction
  __shared__ float sO[MAPPX];   // o tile
  const int b = blockIdx.x;
  const int t = threadIdx.x;
  const float* segb = seg + (size_t)b * MAPPX;
  const float* cb   = c   + (size_t)b * MAPPX;
  const float* ob   = o   + (size_t)b * MAPPX;

  // ---- async DMA the three 16KB tiles into LDS ----
  const unsigned la = lds_off_of(sA);
  const unsigned lc = lds_off_of(sC);
  const unsigned lo = lds_off_of(sO);
#pragma unroll
  for (int j = 0; j < 4; ++j) {
    unsigned byte = (unsigned)(t + j * 256) * 16u;   // 256 lanes x 16B x 4
    async_b128(la + byte, segb, byte);
    async_b128(lc + byte, cb,   byte);
    async_b128(lo + byte, ob,   byte);
  }
  wait_async0();
  __syncthreads();

  // ---- round half-to-even (matches jnp.round), in place ----
#pragma unroll
  for (int k = 0; k < 16; ++k) {
    int i = t + k * 256;
    sA[i] = rintf(sA[i]);
  }
  __syncthreads();

  // ---- dilate pass 1: horizontal max, radius 2 ----
#pragma unroll
  for (int k = 0; k < 16; ++k) {
    int i = t + k * 256, y = i >> 6, x = i & 63;
    int x0 = (x - 2 < 0) ? 0 : x - 2;
    int x1 = (x + 2 > 63) ? 63 : x + 2;
    float m = -__builtin_inff();
    for (int xx = x0; xx <= x1; ++xx) m = fmaxf(m, sA[(y << 6) + xx]);
    sH[i] = m;
  }
  __syncthreads();

  // ---- dilate pass 2 (vertical) -> segment; cm; x8 upsample store ----
#pragma unroll
  for (int k = 0; k < 16; ++k) {
    int i = t + k * 256, y = i >> 6, x = i & 63;
    int y0 = (y - 2 < 0) ? 0 : y - 2;
    int y1 = (y + 2 > 63) ? 63 : y + 2;
    float m = -__builtin_inff();
    for (int yy = y0; yy <= y1; ++yy) m = fmaxf(m, sH[(yy << 6) + x]);
    float cm = sC[i] * m;
    sA[i] = cm;                                    // reuse A buffer as cm
    float4 v4 = make_float4(m, m, m, m);
    float* dst = out + OFF_SEGBIG + (size_t)b * IMG + (size_t)(y * 8) * WF + x * 8;
#pragma unroll
    for (int r = 0; r < 8; ++r) {
      *(float4*)(dst + (size_t)r * WF)     = v4;
      *(float4*)(dst + (size_t)r * WF + 4) = v4;
    }
  }
  __syncthreads();

  // ---- NMS pass 1: horizontal max of cm, radius 7 ----
#pragma unroll
  for (int k = 0; k < 16; ++k) {
    int i = t + k * 256, y = i >> 6, x = i & 63;
    int x0 = (x - 7 < 0) ? 0 : x - 7;
    int x1 = (x + 7 > 63) ? 63 : x + 7;
    float m = -__builtin_inff();
    for (int xx = x0; xx <= x1; ++xx) m = fmaxf(m, sA[(y << 6) + xx]);
    sH[i] = m;
  }
  __syncthreads();

  // ---- NMS pass 2 (vertical) -> keep mask, masked outputs, count ----
  int cnt = 0;
#pragma unroll
  for (int k = 0; k < 16; ++k) {
    int i = t + k * 256, y = i >> 6, x = i & 63;
    int y0 = (y - 7 < 0) ? 0 : y - 7;
    int y1 = (y + 7 > 63) ? 63 : y + 7;
    float m = -__builtin_inff();
    for (int yy = y0; yy <= y1; ++yy) m = fmaxf(m, sH[(yy << 6) + x]);
    float cm = sA[i];
    bool keep = (cm == m) && (cm > THRESH);
    out[OFF_CO + (size_t)b * MAPPX + i] = keep ? sO[i] : 0.0f;
    out[OFF_CC + (size_t)b * MAPPX + i] = keep ? cm    : 0.0f;
    cnt += keep ? 1 : 0;
  }
  __syncthreads();
  sC[t] = (float)cnt;              // sC tile no longer needed
  __syncthreads();
  for (int s = 128; s > 0; s >>= 1) {
    if (t < s) sC[t] += sC[t + s];
    __syncthreads();
  }
  if (t == 0) out[OFF_MNT + b] = sC[0];   // int32 in ref; concat promotes to f32
}

// ---------------- kernel 4: confidence = concat(c,w,h,o), vectorized ----------------
__global__ __launch_bounds__(256) void kconcat(const float* __restrict__ c,
                                               const float* __restrict__ w,
                                               const float* __restrict__ h,
                                               const float* __restrict__ o,
                                               float* __restrict__ out) {
  const int i4   = blockIdx.x * 256 + threadIdx.x;  // < 131072 float4
  const int b    = i4 >> 12;                        // 4096 float4 per image
  const int ch   = (i4 >> 10) & 3;                  // 1024 float4 per channel
  const int pix4 = i4 & 1023;
  const float* src = (ch == 0) ? c : (ch == 1) ? w : (ch == 2) ? h : o;
  ((float4*)out)[i4] = ((const float4*)(src + (size_t)b * MAPPX))[pix4];
}

// ---------------- launch ----------------
extern "C" void kernel_launch(void* const* d_in, const int* in_sizes, int n_in,
                              void* d_out, int out_size, void* d_ws, size_t ws_size,
                              hipStream_t stream) {
  (void)in_sizes; (void)n_in; (void)out_size; (void)ws_size;
  // setup_inputs order: imgs, enhanceImageP, c, w, h, o, seg, ori
  const float* enh = (const float*)d_in[1];
  const float* c   = (const float*)d_in[2];
  const float* w   = (const float*)d_in[3];
  const float* h   = (const float*)d_in[4];
  const float* o   = (const float*)d_in[5];
  const float* seg = (const float*)d_in[6];
  float* out = (float*)d_out;
  float* mm  = (float*)d_ws;     // 64 floats: [min x32, max x32]

  kminmax<<<B_IMG, 1024, 0, stream>>>(enh, mm);
  knorm  <<<(B_IMG * IMG) / (256 * 4), 256, 0, stream>>>(enh, mm, out + OFF_ENH);
  kmaps  <<<B_IMG, 256, 0, stream>>>(seg, c, o, out);
  kconcat<<<(B_IMG * 4 * MAPPX) / (256 * 4), 256, 0, stream>>>(c, w, h, o, out + OFF_CONF);
}